// MultiRE_30030411334074
// MI455X (gfx1250) — compile-verified
//
#include <hip/hip_runtime.h>
#include <hip/hip_bf16.h>
#include <math.h>

#define NDIM 230
#define DPAD 232      // NDIM padded to multiple of 4 (K step)
#define NREL 176
#define NBAG 160
#define MAXR 32       // max sentences per bag (<=30), padded
#define NTIL 11       // 176 / 16 column tiles
#define KSTP 58       // DPAD / 4 WMMA K-steps (last one peeled)

typedef __attribute__((ext_vector_type(2))) float v2f;
typedef __attribute__((ext_vector_type(8))) float v8f;

// dynamic shared memory layout (floats)
#define OFF_X    0
#define OFF_E    (OFF_X + MAXR * DPAD)     // a then exp(a-m)
#define OFF_DEN  (OFF_E + MAXR * NREL)
#define OFF_S    (OFF_DEN + NREL)
#define OFF_L    (OFF_S + 16 * DPAD)
#define OFF_META (OFF_L + 16 * NREL)
#define SHM_FLOATS (OFF_META + 8)

__global__ __launch_bounds__(256)
void multire_bag_kernel(const float* __restrict__ inp_en,
                        const float* __restrict__ inp_zh,
                        const float* __restrict__ rel_emb,
                        const float* __restrict__ M_w,
                        const float* __restrict__ M_b,
                        const int*   __restrict__ l_en,
                        const int*   __restrict__ l_zh,
                        float*       __restrict__ out)
{
    extern __shared__ float smem[];
    float* X   = smem + OFF_X;     // [32][DPAD] bag sentence vectors, zero padded
    float* E   = smem + OFF_E;     // [32][NREL] scores -> exp weights
    float* DEN = smem + OFF_DEN;   // [NREL] softmax denominators
    float* S   = smem + OFF_S;     // [16][DPAD] current S tile
    float* L   = smem + OFF_L;     // [16][NREL] current logits tile
    int*   META = (int*)(smem + OFF_META);

    const int bag  = blockIdx.x;
    const int tid  = threadIdx.x;
    const int lane = tid & 31;
    const int wave = tid >> 5;
    const int lh   = lane >> 4;    // lane half (0/1)
    const int ll   = lane & 15;

    if (tid == 0) {
        int se = 0, sz = 0;
        for (int j = 0; j < bag; ++j) { se += l_en[j]; sz += l_zh[j]; }
        META[0] = se; META[1] = sz; META[2] = l_en[bag]; META[3] = l_zh[bag];
    }
    __syncthreads();
    const int start_en = META[0], start_zh = META[1];
    const int n_en = META[2], n_zh = META[3];
    const int n = n_en + n_zh;           // joint en+zh sentence count (<=30)

    // ---- load bag sentences into LDS, zero padded to 32 x DPAD ----
    for (int idx = tid; idx < MAXR * DPAD; idx += blockDim.x) {
        const int r = idx / DPAD, c = idx - r * DPAD;
        float v = 0.0f;
        if (c < NDIM) {
            if (r < n_en)   v = inp_en[(start_en + r) * NDIM + c];
            else if (r < n) v = inp_zh[(start_zh + (r - n_en)) * NDIM + c];
        }
        X[idx] = v;
    }
    __syncthreads();

    // ---- Phase 1: a[t][k] = X @ rel_emb^T  (f32 WMMA, 2 Mtiles x 11 Ntiles) ----
    for (int tile = wave; tile < 2 * NTIL; tile += 8) {
        const int mt = tile / NTIL;
        const int nt = tile - mt * NTIL;
        const int mrow = mt * 16 + ll;       // A row held by this lane
        const int ncol = nt * 16 + ll;       // B col held by this lane
        const float* ap = X + mrow * DPAD + 2 * lh;          // 8B aligned
        const float* bp = rel_emb + ncol * NDIM + 2 * lh;    // 8B aligned
        v8f acc = {0.f,0.f,0.f,0.f,0.f,0.f,0.f,0.f};
        for (int kk = 0; kk < KSTP - 1; ++kk) {              // K = 0..227, no guards
            v2f a = *(const v2f*)ap;
            v2f b = *(const v2f*)bp;
            acc = __builtin_amdgcn_wmma_f32_16x16x4_f32(false, a, false, b,
                                                        (short)0, acc, false, false);
            ap += 4; bp += 4;
        }
        {   // peeled K-step 57: K = 228..231 (A cols 230/231 are zero in LDS)
            v2f a = *(const v2f*)ap;
            v2f b = {0.f, 0.f};
            if (lh == 0) b = *(const v2f*)bp;                // K=228,229 valid
            acc = __builtin_amdgcn_wmma_f32_16x16x4_f32(false, a, false, b,
                                                        (short)0, acc, false, false);
        }
        // D layout: lane holds N=ll, rows M = j + 8*lh
        #pragma unroll
        for (int j = 0; j < 8; ++j)
            E[(mt * 16 + j + 8 * lh) * NREL + nt * 16 + ll] = acc[j];
    }
    __syncthreads();

    // ---- Phase 2: joint ragged softmax over the n sentences, per relation ----
    for (int k = tid; k < NREL; k += blockDim.x) {
        float m = -3.0e38f;
        for (int t = 0; t < n; ++t) m = fmaxf(m, E[t * NREL + k]);
        float s = 0.0f;
        for (int t = 0; t < n; ++t) {
            float e = __expf(E[t * NREL + k] - m);
            E[t * NREL + k] = e;
            s += e;
        }
        DEN[k] = s;
    }
    __syncthreads();

    // ---- Phase 3: per 16-relation tile: S tile -> GEMM2 -> diag log-softmax ----
    for (int kt = 0; kt < NTIL; ++kt) {
        // S[kr][c] = sum_t e[t][k] * X[t][c] / denom[k]
        for (int idx = tid; idx < 16 * DPAD; idx += blockDim.x) {
            const int kr = idx / DPAD, c = idx - kr * DPAD;
            float accv = 0.0f;
            if (c < NDIM) {
                const int k = kt * 16 + kr;
                for (int t = 0; t < n; ++t)
                    accv += E[t * NREL + k] * X[t * DPAD + c];
                accv /= DEN[k];
            }
            S[idx] = accv;
        }
        __syncthreads();

        // L[16][176] = S @ M_w^T + M_b  (f32 WMMA, 11 Ntiles over 8 waves)
        v8f accT[2];
        int nts[2];
        int cnt = 0;
        for (int nt = wave; nt < NTIL; nt += 8) {
            const int ncol = nt * 16 + ll;
            const float* ap = S + ll * DPAD + 2 * lh;
            const float* bp = M_w + ncol * NDIM + 2 * lh;
            v8f acc = {0.f,0.f,0.f,0.f,0.f,0.f,0.f,0.f};
            for (int kk = 0; kk < KSTP - 1; ++kk) {
                v2f a = *(const v2f*)ap;
                v2f b = *(const v2f*)bp;
                acc = __builtin_amdgcn_wmma_f32_16x16x4_f32(false, a, false, b,
                                                            (short)0, acc, false, false);
                ap += 4; bp += 4;
            }
            {   // peeled boundary K-step
                v2f a = *(const v2f*)ap;
                v2f b = {0.f, 0.f};
                if (lh == 0) b = *(const v2f*)bp;
                acc = __builtin_amdgcn_wmma_f32_16x16x4_f32(false, a, false, b,
                                                            (short)0, acc, false, false);
            }
            accT[cnt] = acc; nts[cnt] = nt; ++cnt;
        }
        __syncthreads();                       // all waves done reading S
        for (int q = 0; q < cnt; ++q) {
            const int nt = nts[q];
            const float bias = M_b[nt * 16 + ll];
            #pragma unroll
            for (int j = 0; j < 8; ++j)
                L[(j + 8 * lh) * NREL + nt * 16 + ll] = accT[q][j] + bias;
        }
        __syncthreads();

        // out[bag][kt*16+m] = L[m][kt*16+m] - logsumexp_j L[m][j]
        if (tid < 16) {
            const int kg = kt * 16 + tid;
            float m = -3.0e38f;
            for (int j = 0; j < NREL; ++j) m = fmaxf(m, L[tid * NREL + j]);
            float s = 0.0f;
            for (int j = 0; j < NREL; ++j) s += __expf(L[tid * NREL + j] - m);
            out[bag * NREL + kg] = L[tid * NREL + kg] - m - __logf(s);
        }
        __syncthreads();                       // S / L reused next kt
    }
}

extern "C" void kernel_launch(void* const* d_in, const int* in_sizes, int n_in,
                              void* d_out, int out_size, void* d_ws, size_t ws_size,
                              hipStream_t stream) {
    (void)in_sizes; (void)n_in; (void)out_size; (void)d_ws; (void)ws_size;
    const float* inp_en  = (const float*)d_in[0];
    const float* inp_zh  = (const float*)d_in[1];
    const float* rel_emb = (const float*)d_in[2];
    const float* M_w     = (const float*)d_in[3];
    const float* M_b     = (const float*)d_in[4];
    // d_in[5] = r_en (identity relation ids, unused)
    const int*   l_en    = (const int*)d_in[6];
    // d_in[7] = r_zh (unused)
    const int*   l_zh    = (const int*)d_in[8];
    // d_in[9] = re_mask (identity diagonal, unused)
    float* out = (float*)d_out;

    const size_t shm = SHM_FLOATS * sizeof(float);   // ~79 KB < 320 KB LDS
    (void)hipFuncSetAttribute((const void*)multire_bag_kernel,
                              hipFuncAttributeMaxDynamicSharedMemorySize, (int)shm);
    multire_bag_kernel<<<dim3(NBAG), dim3(256), shm, stream>>>(
        inp_en, inp_zh, rel_emb, M_w, M_b, l_en, l_zh, out);
}